// QuantumLayer_33036888441061
// MI455X (gfx1250) — compile-verified
//
#include <hip/hip_runtime.h>
#include <math.h>

// Quantum layer simulator for MI455X (gfx1250).
// state (1024 complex) kept in LDS per wave; each layer's 10 one-qubit gates
// are fused into two dense 32x32 complex GEMMs done with V_WMMA_F32_16X16X4_F32
// (full fp32 precision); the CNOT chain is a Gray-code index permutation.
// Per-layer gate matrices are staged into LDS with GLOBAL_LOAD_ASYNC_TO_LDS_B128
// (ASYNCcnt path).

#define NQ      10
#define NL      4
#define BATCH   2048
#define PITCH   36            // row pitch: 144B -> 16B aligned rows, conflict-free A loads
#define PLANE   (32 * PITCH)  // 1152 floats per 32x32 plane

typedef float v2f __attribute__((ext_vector_type(2)));
typedef float v8f __attribute__((ext_vector_type(8)));

__device__ __forceinline__ float2 cmulc(float2 a, float2 b) {
    return make_float2(a.x * b.x - a.y * b.y, a.x * b.y + a.y * b.x);
}

__device__ __forceinline__ v8f wmma4(v2f a, v2f b, v8f c) {
    // D(16x16,f32) = A(16x4,f32) * B(4x16,f32) + C
    return __builtin_amdgcn_wmma_f32_16x16x4_f32(
        /*neg_a=*/false, a, /*neg_b=*/false, b,
        /*c_mod=*/(short)0, c, /*reuse_a=*/false, /*reuse_b=*/false);
}

// ---------------------------------------------------------------------------
// Precompute kernel: build A[l] = U(l,0)x..xU(l,4) and B[l] = U(l,5)x..xU(l,9)
// (32x32 complex each) from q_params.  Layout in d_ws (floats):
//   per layer l: [Ar 1024][Ai 1024][Br 1024][Bi 1024]  (row-major 32x32)
// ---------------------------------------------------------------------------
__global__ __launch_bounds__(256) void qprep_kernel(const float* __restrict__ qp,
                                                    float* __restrict__ AB) {
    int t = blockIdx.x * 256 + threadIdx.x;
    if (t >= NL * 2 * 1024) return;
    int l    = t >> 11;
    int half = (t >> 10) & 1;   // 0 -> A (qubits 0..4), 1 -> B (qubits 5..9)
    int e    = t & 1023;
    int r = e >> 5, c = e & 31;

    float2 acc = make_float2(1.0f, 0.0f);
#pragma unroll
    for (int q = 0; q < 5; ++q) {
        int n = half * 5 + q;
        const float* w = qp + (l * NQ + n) * 3;
        float sx, cx, sy, cy, sz, cz;
        sincosf(0.5f * w[0], &sx, &cx);
        sincosf(0.5f * w[1], &sy, &cy);
        sincosf(0.5f * w[2], &sz, &cz);
        // M = Ury * Urx
        float2 m00 = make_float2( cy * cx,  sy * sx);
        float2 m01 = make_float2(-sy * cx, -cy * sx);
        float2 m10 = make_float2( sy * cx, -cy * sx);
        float2 m11 = make_float2( cy * cx, -sy * sx);
        float2 eN = make_float2(cz, -sz);   // exp(-i w2/2)
        float2 eP = make_float2(cz,  sz);   // conj
        float2 u00 = cmulc(eN, m00), u01 = cmulc(eN, m01);
        float2 u10 = cmulc(eP, m10), u11 = cmulc(eP, m11);
        int br = (r >> (4 - q)) & 1;
        int bc = (c >> (4 - q)) & 1;
        float2 u = br ? (bc ? u11 : u10) : (bc ? u01 : u00);
        acc = cmulc(acc, u);
    }
    int base = l * 4096 + half * 2048;
    AB[base + e]        = acc.x;
    AB[base + 1024 + e] = acc.y;
}

// ---------------------------------------------------------------------------
// Complex 32x32 GEMM in LDS via WMMA f32 16x16x4.
//   D = Aop * Bop, complex.  TRANSB: Bop(k,n) = Bplane[n][k] (for S*B^T).
// All planes are 32 rows x PITCH floats in LDS.
// ---------------------------------------------------------------------------
template <bool TRANSB>
__device__ __forceinline__ void cgemm32(const float* Ar, const float* Ai,
                                        const float* Br_, const float* Bi_,
                                        float* Dr, float* Di, int lane) {
    const int m    = lane & 15;
    const int half = lane >> 4;
#pragma unroll 1
    for (int IJ = 0; IJ < 4; ++IJ) {
        const int I = (IJ >> 1) * 16;   // output row tile
        const int J = (IJ & 1) * 16;    // output col tile
        v8f dr = {0.f, 0.f, 0.f, 0.f, 0.f, 0.f, 0.f, 0.f};
        v8f di = {0.f, 0.f, 0.f, 0.f, 0.f, 0.f, 0.f, 0.f};
#pragma unroll
        for (int k0 = 0; k0 < 32; k0 += 4) {
            const int ka = k0 + 2 * half;  // lanes 0-15: k0,k0+1; 16-31: k0+2,k0+3
            v2f ar, ai, br, bi;
            {
                const float* p = Ar + (I + m) * PITCH + ka;
                ar.x = p[0]; ar.y = p[1];
                const float* q = Ai + (I + m) * PITCH + ka;
                ai.x = q[0]; ai.y = q[1];
            }
            if (TRANSB) {
                const float* p = Br_ + (J + m) * PITCH + ka;
                br.x = p[0]; br.y = p[1];
                const float* q = Bi_ + (J + m) * PITCH + ka;
                bi.x = q[0]; bi.y = q[1];
            } else {
                br.x = Br_[ka * PITCH + J + m];
                br.y = Br_[(ka + 1) * PITCH + J + m];
                bi.x = Bi_[ka * PITCH + J + m];
                bi.y = Bi_[(ka + 1) * PITCH + J + m];
            }
            v2f nai; nai.x = -ai.x; nai.y = -ai.y;
            dr = wmma4(ar,  br, dr);   // + ArBr
            dr = wmma4(nai, bi, dr);   // - AiBi
            di = wmma4(ar,  bi, di);   // + ArBi
            di = wmma4(ai,  br, di);   // + AiBr
        }
#pragma unroll
        for (int r = 0; r < 8; ++r) {
            int row = I + r + 8 * half;
            Dr[row * PITCH + J + m] = dr[r];
            Di[row * PITCH + J + m] = di[r];
        }
    }
}

// ---------------------------------------------------------------------------
// Main kernel: one wave per batch element, 2 waves per block.
// ---------------------------------------------------------------------------
__global__ __launch_bounds__(64) void qsim_kernel(const float* __restrict__ x,
                                                  const float* __restrict__ AB,
                                                  float* __restrict__ out) {
    __shared__ float smem[12 * PLANE];  // [Ar Ai Br Bi][wave0: Sr Si Tr Ti][wave1: ...]
    const int lane = threadIdx.x & 31;
    const int wid  = threadIdx.x >> 5;
    const int b    = blockIdx.x * 2 + wid;

    float* sAr = smem;
    float* sAi = smem + PLANE;
    float* sBr = smem + 2 * PLANE;
    float* sBi = smem + 3 * PLANE;
    float* wb  = smem + 4 * PLANE + wid * 4 * PLANE;
    float* Sr = wb;             float* Si = wb + PLANE;
    float* Tr = wb + 2 * PLANE; float* Ti = wb + 3 * PLANE;

    // LDS byte offset of smem[0] (addrspace(3) offsets live in the low 32 bits)
    const unsigned lds_base = (unsigned)(uintptr_t)(&smem[0]);

    // ---- initial product state: amp(idx) = prod_q (bit ? sin : cos)(x_q/2) ----
    float cv[10], sv[10];
#pragma unroll
    for (int q = 0; q < NQ; ++q) {
        float h = 0.5f * x[b * NQ + q];
        cv[q] = cosf(h);
        sv[q] = sinf(h);
    }
#pragma unroll 1
    for (int t = lane; t < 1024; t += 32) {
        float prod = 1.0f;
#pragma unroll
        for (int q = 0; q < NQ; ++q)
            prod *= ((t >> (9 - q)) & 1) ? sv[q] : cv[q];
        int adr = (t >> 5) * PITCH + (t & 31);
        Sr[adr] = prod;
        Si[adr] = 0.0f;
    }

#pragma unroll 1
    for (int l = 0; l < NL; ++l) {
        __syncthreads();  // previous layer's reads of sA/sB done
        // Stage this layer's A/B (re/im planes, 16KB) into LDS via async DMA:
        // 1024 x B128 transfers (16B each), 64 threads -> 16 issues per thread.
#pragma unroll 1
        for (int g = threadIdx.x; g < 1024; g += 64) {
            int plane = g >> 8;
            int idx   = (g & 255) * 4;           // first of 4 consecutive floats
            int r     = idx >> 5;
            int c     = idx & 31;
            unsigned la = lds_base + (unsigned)((plane * PLANE + r * PITCH + c) * 4);
            const float* gp = AB + (size_t)l * 4096 + (size_t)g * 4;
            asm volatile("global_load_async_to_lds_b128 %0, %1, off"
                         :: "v"(la), "v"(gp) : "memory");
        }
        asm volatile("s_wait_asynccnt 0x0" ::: "memory");
        __syncthreads();

        // T = A * S          (row-group gates, qubits 0..4)
        cgemm32<false>(sAr, sAi, Sr, Si, Tr, Ti, lane);
        __builtin_amdgcn_wave_barrier();
        // S = T * B^T        (col-group gates, qubits 5..9)
        cgemm32<true>(Tr, Ti, sBr, sBi, Sr, Si, lane);
        __builtin_amdgcn_wave_barrier();

        // CNOT chain 0->1,...,8->9  ==  Gray-code gather: new[j] = old[j ^ (j>>1)]
#pragma unroll 1
        for (int t = lane; t < 1024; t += 32) {
            int s2 = t ^ (t >> 1);
            int da = (t >> 5) * PITCH + (t & 31);
            int sa = (s2 >> 5) * PITCH + (s2 & 31);
            Tr[da] = Sr[sa];
            Ti[da] = Si[sa];
        }
        __builtin_amdgcn_wave_barrier();
        float* tp;
        tp = Sr; Sr = Tr; Tr = tp;
        tp = Si; Si = Ti; Ti = tp;
    }

    // ---- readout: ev_q = sum_idx |amp|^2 * (1 - 2*bit_q(idx)) ----
    float ev[10];
#pragma unroll
    for (int q = 0; q < NQ; ++q) ev[q] = 0.0f;
#pragma unroll 1
    for (int t = lane; t < 1024; t += 32) {
        int adr  = (t >> 5) * PITCH + (t & 31);
        float re = Sr[adr], im = Si[adr];
        float p  = re * re + im * im;
#pragma unroll
        for (int q = 0; q < NQ; ++q)
            ev[q] += ((t >> (9 - q)) & 1) ? -p : p;
    }
#pragma unroll
    for (int q = 0; q < NQ; ++q) {
        float v = ev[q];
        v += __shfl_xor(v, 16, 32);
        v += __shfl_xor(v, 8, 32);
        v += __shfl_xor(v, 4, 32);
        v += __shfl_xor(v, 2, 32);
        v += __shfl_xor(v, 1, 32);
        if (lane == q) out[b * NQ + q] = v;
    }
}

extern "C" void kernel_launch(void* const* d_in, const int* in_sizes, int n_in,
                              void* d_out, int out_size, void* d_ws, size_t ws_size,
                              hipStream_t stream) {
    const float* x  = (const float*)d_in[0];   // (2048, 10) f32
    const float* qp = (const float*)d_in[1];   // (120,) f32
    float* out = (float*)d_out;                // (2048, 10) f32
    float* AB  = (float*)d_ws;                 // needs 16384 floats (64 KB)

    qprep_kernel<<<32, 256, 0, stream>>>(qp, AB);
    qsim_kernel<<<BATCH / 2, 64, 0, stream>>>(x, AB, out);

    (void)in_sizes; (void)n_in; (void)out_size; (void)ws_size;
}